// AdaptiveFrequencyModulation_77395310674222
// MI455X (gfx1250) — compile-verified
//
#include <hip/hip_runtime.h>
#include <stdint.h>

// ---------------------------------------------------------------------------
// AdaptiveFrequencyModulation for MI455X (gfx1250, wave32)
//   approx band : segmented 4x8-bit stable LSD radix sort (24 segments of 1M)
//                 with LDS-staged coalesced scatter, TDM-staged pivot table,
//                 binary-search rank + phase factor
//   detail bands: deterministic 2-stage reduction + elementwise scale
// ---------------------------------------------------------------------------

#define HN        (1024u * 1024u)          // elements per (b,c) slice
#define NT        (12u * HN)               // elements per tensor  (4*3*1024*1024)
#define NSEG      24u                      // 12 content + 12 style segments
#define RTHREADS  256u
#define RE        16                       // elements per thread per radix tile
#define RTILE     (RTHREADS * RE)          // 4096
#define RBPS      (HN / RTILE)             // 256 radix blocks per segment
#define RGRID     (NSEG * RBPS)            // 6144
#define SCAN_LEN  (256u * RBPS)            // 65536 histogram entries per segment

#if defined(__has_builtin)
#  if __has_builtin(__builtin_amdgcn_tensor_load_to_lds) && __has_builtin(__builtin_amdgcn_s_wait_tensorcnt)
#    define USE_TDM 1
#  endif
#endif
#ifndef USE_TDM
#  define USE_TDM 0
#endif

typedef unsigned int u32x4 __attribute__((ext_vector_type(4)));
typedef int          i32x8 __attribute__((ext_vector_type(8)));
typedef int          i32x4 __attribute__((ext_vector_type(4)));

static __device__ __forceinline__ unsigned ballot32(bool p) {
#if defined(__has_builtin)
#  if __has_builtin(__builtin_amdgcn_ballot_w32)
  return __builtin_amdgcn_ballot_w32(p);
#  else
  return (unsigned)__ballot((int)p);
#  endif
#else
  return (unsigned)__ballot((int)p);
#endif
}

// |x| as orderable unsigned bits (non-negative f32: uint order == float order)
static __device__ __forceinline__ uint32_t abskey(float x) {
  return __float_as_uint(x) & 0x7FFFFFFFu;
}

// ---------------------------------------------------------------------------
// Radix pass 1/3: per-(segment,block) 256-bin histogram.
// Layout gHist[(s*256 + digit)*RBPS + block]  (digit-major per segment)
// ---------------------------------------------------------------------------
__global__ void radix_hist(const uint32_t* __restrict__ keysIn,
                           const float* __restrict__ capx,
                           const float* __restrict__ sapx,
                           int shift, int pass0,
                           uint32_t* __restrict__ gHist) {
  __shared__ uint32_t h[256];
  const unsigned t = threadIdx.x;
  const unsigned s = blockIdx.x / RBPS;
  const unsigned b = blockIdx.x % RBPS;
  h[t] = 0;
  __syncthreads();
  const unsigned base = b * RTILE;
  const float* srcf = (s < 12u) ? (capx + (size_t)s * HN) : (sapx + (size_t)(s - 12u) * HN);
  const uint32_t* seg = keysIn + (size_t)s * HN;
  for (int i = 0; i < RE; ++i) {
    const unsigned off = base + (unsigned)i * RTHREADS + t;
    const uint32_t key = pass0 ? abskey(srcf[off]) : seg[off];
    atomicAdd(&h[(key >> shift) & 0xFFu], 1u);
  }
  __syncthreads();
  gHist[((size_t)s * 256u + t) * RBPS + b] = h[t];
}

// ---------------------------------------------------------------------------
// Radix pass 2/3: per-segment exclusive scan of 65536 entries (in-place).
// One 1024-thread workgroup per segment; 64 entries per thread.
// ---------------------------------------------------------------------------
__global__ void radix_scan(uint32_t* __restrict__ gHist) {
  __shared__ uint32_t ssum[1024];
  const unsigned t = threadIdx.x;
  uint32_t* p = gHist + (size_t)blockIdx.x * SCAN_LEN;
  const unsigned base = t * 64u;
  uint32_t sum = 0;
  for (int i = 0; i < 64; ++i) sum += p[base + i];
  ssum[t] = sum;
  __syncthreads();
  for (unsigned off = 1; off < 1024; off <<= 1) {
    uint32_t v = (t >= off) ? ssum[t - off] : 0u;
    __syncthreads();
    ssum[t] += v;
    __syncthreads();
  }
  uint32_t run = ssum[t] - sum;  // exclusive prefix for this chunk
  for (int i = 0; i < 64; ++i) { uint32_t v = p[base + i]; p[base + i] = run; run += v; }
}

// ---------------------------------------------------------------------------
// Radix pass 3/3: stable scatter with LDS staging.
//  - Wave w owns a contiguous 32*RE chunk read wave-striped, so rank order
//    (w, e, lane) == memory order -> stable pass.
//  - Ranking via wave32 ballot multisplit into per-wave LDS histograms.
//  - Keys are first placed in block-sorted order in LDS, then written out in
//    that order: same-digit elements form contiguous global runs -> coalesced
//    HBM stores instead of random 4B scatter.
// ---------------------------------------------------------------------------
__global__ void radix_scatter(const uint32_t* __restrict__ keysIn,
                              const float* __restrict__ capx,
                              const float* __restrict__ sapx,
                              int shift, int pass0,
                              const uint32_t* __restrict__ gHist,
                              uint32_t* __restrict__ keysOut) {
  __shared__ uint32_t wh[8 * 256];    // per-wave digit counts
  __shared__ uint32_t wb[8 * 256];    // cross-wave exclusive offsets
  __shared__ uint32_t db[256];        // scanned global base for (s, d, b)
  __shared__ uint32_t tot[256];       // block digit totals (scanned in place)
  __shared__ uint32_t lbase[256];     // block-local exclusive digit base
  __shared__ int      adj[256];       // db[d] - lbase[d]
  __shared__ uint32_t stage[RTILE];   // block-sorted keys
  const unsigned t = threadIdx.x, lane = t & 31u, w = t >> 5;
  const unsigned s = blockIdx.x / RBPS;
  const unsigned b = blockIdx.x % RBPS;
  for (unsigned i = t; i < 8u * 256u; i += RTHREADS) wh[i] = 0;
  db[t] = gHist[((size_t)s * 256u + t) * RBPS + b];
  __syncthreads();

  const unsigned chunk = b * RTILE + w * (32u * RE);
  const float* srcf = (s < 12u) ? (capx + (size_t)s * HN) : (sapx + (size_t)(s - 12u) * HN);
  const uint32_t* seg = keysIn + (size_t)s * HN;
  uint32_t keyv[RE];
  uint32_t rankv[RE];
#pragma unroll
  for (int e = 0; e < RE; ++e) {
    const unsigned off = chunk + (unsigned)e * 32u + lane;
    const uint32_t key = pass0 ? abskey(srcf[off]) : seg[off];
    const unsigned d = (key >> shift) & 0xFFu;
    unsigned mask = 0xFFFFFFFFu;
#pragma unroll
    for (int bb = 0; bb < 8; ++bb) {
      const unsigned bit = (d >> bb) & 1u;
      const unsigned bal = ballot32(bit != 0u);
      mask &= bit ? bal : ~bal;
    }
    const unsigned r   = __popc(mask & ((1u << lane) - 1u));
    const unsigned old = wh[w * 256u + d];              // all peers read same value
    if (r == 0) wh[w * 256u + d] = old + __popc(mask);  // group leader updates
    keyv[e]  = key;
    rankv[e] = old + r;
  }
  __syncthreads();

  // cross-wave exclusive scan per digit (thread t handles digit t)
  {
    unsigned run = 0;
#pragma unroll
    for (int ww = 0; ww < 8; ++ww) {
      wb[(unsigned)ww * 256u + t] = run;
      run += wh[(unsigned)ww * 256u + t];
    }
    tot[t] = run;  // block total for digit t
  }
  __syncthreads();

  // exclusive scan of block digit totals (256 entries, Hillis-Steele in LDS)
  const unsigned mytot = tot[t];
  for (unsigned off = 1; off < 256; off <<= 1) {
    const uint32_t v = (t >= off) ? tot[t - off] : 0u;
    __syncthreads();
    tot[t] += v;
    __syncthreads();
  }
  lbase[t] = tot[t] - mytot;
  adj[t]   = (int)db[t] - (int)(tot[t] - mytot);
  __syncthreads();

  // place keys into block-sorted order in LDS
#pragma unroll
  for (int e = 0; e < RE; ++e) {
    const unsigned d = (keyv[e] >> shift) & 0xFFu;
    stage[lbase[d] + wb[w * 256u + d] + rankv[e]] = keyv[e];
  }
  __syncthreads();

  // write out in block-sorted order: same-digit runs are contiguous globally
  uint32_t* outSeg = keysOut + (size_t)s * HN;
#pragma unroll
  for (int k = 0; k < RE; ++k) {
    const unsigned j = (unsigned)k * RTHREADS + t;
    const uint32_t key = stage[j];
    const unsigned d = (key >> shift) & 0xFFu;
    outSeg[(unsigned)((int)j + adj[d])] = key;
  }
}

// ---------------------------------------------------------------------------
// Pivot table: pivots[s][k] = sorted_src[s*HN + k*1024], contiguous per seg.
// ---------------------------------------------------------------------------
__global__ void pivot_build(const uint32_t* __restrict__ sorted,
                            uint32_t* __restrict__ pivots) {
  const unsigned idx = blockIdx.x * 256u + threadIdx.x;  // < 12288
  const unsigned s = idx >> 10, k = idx & 1023u;
  pivots[idx] = sorted[(size_t)s * HN + (size_t)k * 1024u];
}

// ---------------------------------------------------------------------------
// Approx band: rank = lower_bound(s_sorted, |c|) (== searchsorted left),
// out = t_sorted[rank] * cos(0.8*angle_c + 0.2*angle_s).
// Pivot table DMA'd to LDS via the Tensor Data Mover (wave 0 only).
// ---------------------------------------------------------------------------
__global__ void approx_map(const float* __restrict__ content,
                           const float* __restrict__ style,
                           const uint32_t* __restrict__ sorted,
                           const uint32_t* __restrict__ pivots,
                           float* __restrict__ out) {
  __shared__ uint32_t spiv[1024];
  const unsigned t = threadIdx.x;
  const unsigned s = blockIdx.x >> 10;
  const unsigned b = blockIdx.x & 1023u;

#if USE_TDM
  if (t < 32u) {  // one wave issues the DMA; TENSOR ops ignore EXEC
    const unsigned lds_off = (unsigned)(uintptr_t)(void*)spiv;  // low 32 bits = LDS offset
    const unsigned long long ga =
        (unsigned long long)(uintptr_t)(pivots + (size_t)s * 1024u);
    // D# group0: count=1 | lds_addr | global_addr[56:0] | type=2
    u32x4 g0 = {1u, lds_off, (unsigned)ga,
                ((unsigned)((ga >> 32) & 0x01FFFFFFull)) | (2u << 30)};
    // D# group1: data_size=4B; tensor_dim0=1024; tensor_dim1=1;
    //            tile_dim0=1024; tile_dim1=1; tensor_dim0_stride=1024
    i32x8 g1 = {(int)(2u << 16),
                (int)(1024u << 16),   // tensor_dim0 low16 in [31:16]
                (int)(1u << 16),      // tensor_dim0 hi=0, tensor_dim1 low16=1
                (int)(1024u << 16),   // tensor_dim1 hi=0, tile_dim0=1024
                (int)1,               // tile_dim1=1, tile_dim2=0
                (int)1024,            // tensor_dim0_stride low32
                0, 0};
    i32x4 z4 = {0, 0, 0, 0};
    i32x8 z8 = {0, 0, 0, 0, 0, 0, 0, 0};
    // amdgpu-toolchain (clang-23) 6-arg form: g0, g1, g2, g3, extra, cpol
    __builtin_amdgcn_tensor_load_to_lds(g0, g1, z4, z4, z8, 0);
    __builtin_amdgcn_s_wait_tensorcnt(0);
  }
#else
  for (unsigned i = t; i < 1024u; i += 256u) spiv[i] = pivots[s * 1024u + i];
#endif
  __syncthreads();

  const uint32_t* S = sorted + (size_t)s * HN;          // sorted |content| slice
  const uint32_t* T = sorted + (size_t)(s + 12u) * HN;  // sorted |style| slice
  const size_t base = (size_t)s * HN + (size_t)b * 1024u + (size_t)t * 4u;

  const float4 c  = *(const float4*)(content + base);
  const float4 st = *(const float4*)(style + base);
  const float cv[4] = {c.x, c.y, c.z, c.w};
  const float sv[4] = {st.x, st.y, st.z, st.w};
  const float tbl[4] = {1.0f, 0.80901699437494745f, -0.80901699437494745f, -1.0f};
  float r[4];
#pragma unroll
  for (int k = 0; k < 4; ++k) {
    const uint32_t cb = __float_as_uint(cv[k]);
    const uint32_t v  = cb & 0x7FFFFFFFu;
    // phase 1: search 1024 LDS pivots (s_sorted sampled at stride 1024)
    unsigned lo = 0, hi = 1024;
    while (lo < hi) { const unsigned m = (lo + hi) >> 1; if (spiv[m] < v) lo = m + 1; else hi = m; }
    const unsigned p  = lo;
    unsigned glo = p ? (p - 1u) * 1024u : 0u;
    unsigned ghi = (p < 1024u) ? p * 1024u + 1u : HN;
    // phase 2: lower_bound inside the ~1K window (L2-resident)
    while (glo < ghi) { const unsigned m = (glo + ghi) >> 1; if (S[m] < v) glo = m + 1; else ghi = m; }
    unsigned rk = glo;
    if (rk >= HN) rk = HN - 1u;
    const uint32_t sb = __float_as_uint(sv[k]);
    r[k] = __uint_as_float(T[rk]) * tbl[((cb >> 31) << 1) | (sb >> 31)];
  }
  *(float4*)(out + base) = make_float4(r[0], r[1], r[2], r[3]);
}

// ---------------------------------------------------------------------------
// Detail bands: deterministic two-stage sum of |c| and |s| -> ratio.
// ---------------------------------------------------------------------------
__global__ void reduce_partial(const float* __restrict__ c1, const float* __restrict__ c2,
                               const float* __restrict__ c3, const float* __restrict__ s1,
                               const float* __restrict__ s2, const float* __restrict__ s3,
                               float* __restrict__ partial) {
  const unsigned band = blockIdx.y;
  const float* c = (band == 0) ? c1 : (band == 1) ? c2 : c3;
  const float* s = (band == 0) ? s1 : (band == 1) ? s2 : s3;
  const unsigned blk = blockIdx.x, t = threadIdx.x;
  const size_t base = (size_t)blk * 12288u + (size_t)t * 4u;
  float sc = 0.f, ss = 0.f;
#pragma unroll
  for (int k = 0; k < 12; ++k) {
    const size_t i = base + (size_t)k * 1024u;
    const float4 a = *(const float4*)(c + i);
    const float4 b = *(const float4*)(s + i);
    sc += fabsf(a.x) + fabsf(a.y) + fabsf(a.z) + fabsf(a.w);
    ss += fabsf(b.x) + fabsf(b.y) + fabsf(b.z) + fabsf(b.w);
  }
  __shared__ float lc[256], ls[256];
  lc[t] = sc; ls[t] = ss;
  __syncthreads();
  for (unsigned off = 128; off; off >>= 1) {
    if (t < off) { lc[t] += lc[t + off]; ls[t] += ls[t + off]; }
    __syncthreads();
  }
  if (t == 0) {
    partial[band * 2048u + blk]         = lc[0];
    partial[band * 2048u + 1024u + blk] = ls[0];
  }
}

__global__ void reduce_final(const float* __restrict__ partial, float* __restrict__ ratios) {
  const unsigned band = blockIdx.x, t = threadIdx.x;
  float sc = 0.f, ss = 0.f;
#pragma unroll
  for (int k = 0; k < 4; ++k) {
    sc += partial[band * 2048u + (unsigned)k * 256u + t];
    ss += partial[band * 2048u + 1024u + (unsigned)k * 256u + t];
  }
  __shared__ float lc[256], ls[256];
  lc[t] = sc; ls[t] = ss;
  __syncthreads();
  for (unsigned off = 128; off; off >>= 1) {
    if (t < off) { lc[t] += lc[t + off]; ls[t] += ls[t + off]; }
    __syncthreads();
  }
  if (t == 0) ratios[band] = (ls[0] / lc[0]) * 1.8f;  // mean ratio * style_scale
}

__global__ void detail_apply(const float* __restrict__ c1, const float* __restrict__ c2,
                             const float* __restrict__ c3, const float* __restrict__ s1,
                             const float* __restrict__ s2, const float* __restrict__ s3,
                             const float* __restrict__ ratios, float* __restrict__ out) {
  const unsigned band = blockIdx.y;
  const float* c = (band == 0) ? c1 : (band == 1) ? c2 : c3;
  const float* s = (band == 0) ? s1 : (band == 1) ? s2 : s3;
  const float ratio = ratios[band];
  float* o = out + (size_t)(band + 1u) * NT;
  const unsigned t = threadIdx.x, bx = blockIdx.x;
  const float tbl[4] = {1.0f, 0.58778525229247312f, -0.58778525229247312f, -1.0f};
#pragma unroll
  for (int k = 0; k < 4; ++k) {
    const size_t i = ((size_t)bx * 1024u + (size_t)k * 256u + t) * 4u;
    const float4 a = *(const float4*)(c + i);
    const float4 b = *(const float4*)(s + i);
    const float av[4] = {a.x, a.y, a.z, a.w};
    const float bv[4] = {b.x, b.y, b.z, b.w};
    float r[4];
#pragma unroll
    for (int j = 0; j < 4; ++j) {
      const unsigned idx = ((__float_as_uint(av[j]) >> 31) << 1) | (__float_as_uint(bv[j]) >> 31);
      r[j] = fabsf(av[j]) * ratio * tbl[idx];
    }
    *(float4*)(o + i) = make_float4(r[0], r[1], r[2], r[3]);
  }
}

// ---------------------------------------------------------------------------
// Host launcher
// ---------------------------------------------------------------------------
extern "C" void kernel_launch(void* const* d_in, const int* in_sizes, int n_in,
                              void* d_out, int out_size, void* d_ws, size_t ws_size,
                              hipStream_t stream) {
  (void)in_sizes; (void)n_in; (void)out_size; (void)ws_size;

  const float* capx = (const float*)d_in[0];  // content_approx
  const float* cdh  = (const float*)d_in[1];
  const float* cdv  = (const float*)d_in[2];
  const float* cdd  = (const float*)d_in[3];
  const float* sapx = (const float*)d_in[4];  // style_approx
  const float* sdh  = (const float*)d_in[5];
  const float* sdv  = (const float*)d_in[6];
  const float* sdd  = (const float*)d_in[7];
  float* out = (float*)d_out;

  // workspace layout
  uint32_t* bufB   = (uint32_t*)d_ws;                 // 24*HN keys (final sorted)
  uint32_t* gHist  = bufB + (size_t)NSEG * HN;        // 24*65536
  uint32_t* pivots = gHist + (size_t)NSEG * SCAN_LEN; // 12*1024
  float*    partial = (float*)(pivots + 12u * 1024u); // 3*2048
  float*    ratios  = partial + 3u * 2048u;           // 3
  // ping buffer A lives in the (not-yet-written) detail region of d_out
  uint32_t* bufA = (uint32_t*)d_out + NT;             // 3*NT uints available >= 24*HN

  // ---- segmented stable LSD radix sort: pass0 in->A, A->B, B->A, A->B ----
  uint32_t* pin[4]  = {nullptr, bufA, bufB, bufA};
  uint32_t* pout[4] = {bufA,   bufB, bufA, bufB};
  for (int pass = 0; pass < 4; ++pass) {
    const int shift = pass * 8;
    const int p0 = (pass == 0) ? 1 : 0;
    radix_hist<<<RGRID, RTHREADS, 0, stream>>>(pin[pass], capx, sapx, shift, p0, gHist);
    radix_scan<<<NSEG, 1024, 0, stream>>>(gHist);
    radix_scatter<<<RGRID, RTHREADS, 0, stream>>>(pin[pass], capx, sapx, shift, p0, gHist, pout[pass]);
  }

  // ---- approx band: pivots + rank mapping ----
  pivot_build<<<48, 256, 0, stream>>>(bufB, pivots);
  approx_map<<<12u * 1024u, 256, 0, stream>>>(capx, sapx, bufB, pivots, out);

  // ---- detail bands (run last: they overwrite the bufA scratch region) ----
  reduce_partial<<<dim3(1024, 3), 256, 0, stream>>>(cdh, cdv, cdd, sdh, sdv, sdd, partial);
  reduce_final<<<3, 256, 0, stream>>>(partial, ratios);
  detail_apply<<<dim3(3072, 3), 256, 0, stream>>>(cdh, cdv, cdd, sdh, sdv, sdd, ratios, out);
}